// AttentionExample_3882650435947
// MI455X (gfx1250) — compile-verified
//
#include <hip/hip_runtime.h>

#define H 4096
#define S 8192
#define W2 (2 * H)

typedef __attribute__((ext_vector_type(2))) float v2f;
typedef __attribute__((ext_vector_type(8))) float v8f;

// ---------------------------------------------------------------------------
// Kernel A: partial v2[c] = sum over a 128-row chunk of other[h] * W[h, H+c]
// grid (4, 32), block 256. Each block: 1024 cols (4/thread, float4), 128 rows.
// ---------------------------------------------------------------------------
__global__ void k_gemv_w(const float* __restrict__ W, const float* __restrict__ other,
                         float* __restrict__ partial) {
  __shared__ float sOther[128];
  const int tid = threadIdx.x;
  const int c0  = blockIdx.x * 1024 + tid * 4;
  const int r0  = blockIdx.y * 128;
  if (tid < 128) sOther[tid] = other[r0 + tid];
  __syncthreads();

  float a0 = 0.f, a1 = 0.f, a2 = 0.f, a3 = 0.f;
  const float* wp = W + (size_t)r0 * W2 + H + c0;
  #pragma unroll 4
  for (int r = 0; r < 128; ++r) {
    const float  o = sOther[r];
    const float4 w = *(const float4*)wp;
    a0 = fmaf(o, w.x, a0);
    a1 = fmaf(o, w.y, a1);
    a2 = fmaf(o, w.z, a2);
    a3 = fmaf(o, w.w, a3);
    wp += W2;
  }
  float* p = partial + (size_t)blockIdx.y * H + c0;
  p[0] = a0; p[1] = a1; p[2] = a2; p[3] = a3;
}

// ---------------------------------------------------------------------------
// Kernel B: v2[c] = sum_{k<32} partial[k*H + c]   (fixed-order, deterministic)
// grid 16, block 256.
// ---------------------------------------------------------------------------
__global__ void k_reduce_v(const float* __restrict__ partial, float* __restrict__ v2) {
  const int c = blockIdx.x * 256 + threadIdx.x;
  float s = 0.f;
  #pragma unroll 8
  for (int k = 0; k < 32; ++k) s += partial[(size_t)k * H + c];
  v2[c] = s;
}

// ---------------------------------------------------------------------------
// Kernel C: scores[s] = enc[s] . v2 via V_WMMA_F32_16X16X4_F32.
// grid 128, block 128 (4 waves). Each wave owns 16 consecutive enc rows.
// A fragment (16x4 f32): lanes 0-15 -> M=lane, VGPR0=K0, VGPR1=K1;
//                        lanes 16-31 -> M=lane-16, VGPR0=K2, VGPR1=K3.
// B fragment (4x16): row k broadcast with v2[c+k] (all N columns identical),
// so every column of D accumulates the same per-row dot product.
// ---------------------------------------------------------------------------
__global__ void k_scores_wmma(const float* __restrict__ enc, const float* __restrict__ v2,
                              float* __restrict__ scores) {
  __shared__ float sv2[H];
  const int tid = threadIdx.x;
  // Stage v2 (16 KB) into LDS, coalesced.
  for (int i = 0; i < H; i += 128) sv2[i + tid] = v2[i + tid];
  __syncthreads();

  const int wave   = tid >> 5;
  const int lane   = tid & 31;
  const int half   = lane >> 4;     // 0 -> K0/K1, 1 -> K2/K3
  const int mrow   = lane & 15;
  const int s_base = (blockIdx.x * 4 + wave) * 16;

  const float* arow = enc + (size_t)(s_base + mrow) * H + 2 * half;

  v8f acc = {};
  for (int c = 0; c < H; c += 4) {
    if ((c & 255) == 0) __builtin_prefetch(arow + c + 256, 0, 1);  // global_prefetch_b8
    const v2f a = *(const v2f*)(arow + c);                          // enc[M, c+2h .. +1]
    const v2f b = *(const v2f*)(&sv2[c + 2 * half]);                // v2[c+2h .. +1] (ds_load_b64)
    acc = __builtin_amdgcn_wmma_f32_16x16x4_f32(
        /*neg_a=*/false, a, /*neg_b=*/false, b,
        /*c_mod=*/(short)0, acc, /*reuse_a=*/false, /*reuse_b=*/false);
  }

  // D layout: VGPR j -> M=j (lanes 0-15), M=8+j (lanes 16-31); all N identical.
  if (mrow == 0) {
    float* sp = scores + s_base + half * 8;
    #pragma unroll
    for (int j = 0; j < 8; ++j) sp[j] = acc[j];
  }
}

// ---------------------------------------------------------------------------
// Kernel D: softmax over 8192 scores. 1 block, 1024 threads, 8 elems/thread.
// ---------------------------------------------------------------------------
__global__ void k_softmax(const float* __restrict__ scores, float* __restrict__ out) {
  __shared__ float red[1024];
  const int tid = threadIdx.x;
  float x[8];
  float mx = -3.402823466e38f;
  #pragma unroll
  for (int i = 0; i < 8; ++i) {
    x[i] = scores[tid + i * 1024];
    mx = fmaxf(mx, x[i]);
  }
  red[tid] = mx;
  __syncthreads();
  for (int s = 512; s > 0; s >>= 1) {
    if (tid < s) red[tid] = fmaxf(red[tid], red[tid + s]);
    __syncthreads();
  }
  const float gmax = red[0];
  __syncthreads();

  float sum = 0.f;
  #pragma unroll
  for (int i = 0; i < 8; ++i) {
    x[i] = __expf(x[i] - gmax);
    sum += x[i];
  }
  red[tid] = sum;
  __syncthreads();
  for (int s = 512; s > 0; s >>= 1) {
    if (tid < s) red[tid] += red[tid + s];
    __syncthreads();
  }
  const float inv = 1.f / red[0];
  #pragma unroll
  for (int i = 0; i < 8; ++i) out[tid + i * 1024] = x[i] * inv;
}

// ---------------------------------------------------------------------------
// Launcher. Inputs: hidden(H), encoder_outputs(S*H), attn_W(H*2H), attn_b(H),
// other(H). hidden/attn_b only contribute a uniform additive constant to the
// scores, which softmax is exactly invariant to -> skipped.
// Workspace: partial (32*H) | v2 (H) | scores (S)  = ~573 KB.
// ---------------------------------------------------------------------------
extern "C" void kernel_launch(void* const* d_in, const int* in_sizes, int n_in,
                              void* d_out, int out_size, void* d_ws, size_t ws_size,
                              hipStream_t stream) {
  (void)in_sizes; (void)n_in; (void)out_size; (void)ws_size;
  const float* enc   = (const float*)d_in[1];
  const float* W     = (const float*)d_in[2];
  const float* other = (const float*)d_in[4];

  float* ws      = (float*)d_ws;
  float* partial = ws;                 // 32*H floats
  float* v2      = ws + 32 * H;        // H floats
  float* scores  = v2 + H;             // S floats

  k_gemv_w<<<dim3(4, 32), 256, 0, stream>>>(W, other, partial);
  k_reduce_v<<<16, 256, 0, stream>>>(partial, v2);
  k_scores_wmma<<<128, 128, 0, stream>>>(enc, v2, scores);
  k_softmax<<<1, 1024, 0, stream>>>(scores, (float*)d_out);
}